// GraphSage_23321672417518
// MI455X (gfx1250) — compile-verified
//
#include <hip/hip_runtime.h>

// GraphSAGE (MI455X / gfx1250, wave32).
// Reference reduces to: out = relu( concat(feats[nodes], mean(feats[neighs])) @ w1.T )
// (layer-0 result is overwritten by the loop; w0 is dead.)

typedef float v2f __attribute__((ext_vector_type(2)));
typedef float v8f __attribute__((ext_vector_type(8)));

#define FEAT   256
#define TWOF   512
#define OUTF   256
#define S_NB   25

#define KT     64      // K slice staged in LDS
#define KTP    66      // padded LDS row stride (floats): 264B rows -> 8B-aligned b64 reads
#define MT     128     // block M tile = 8 waves * 16 rows
#define NT     64      // block N tile = 4 x 16 per wave

// ---------------------------------------------------------------------------
// Kernel 1: X[b, 0:256] = feats[nodes[b]] ; X[b, 256:512] = ragged mean of
// feats[samp_neighs[b, 0:val_lens[b]]].  One block per target node.
// ---------------------------------------------------------------------------
__global__ __launch_bounds__(256)
void sage_prep(const float* __restrict__ feats,
               const int*   __restrict__ nodes,
               const int*   __restrict__ neighs,
               const int*   __restrict__ lens,
               float*       __restrict__ X,
               int B)
{
    __shared__ int sh_idx[S_NB];
    const int b = blockIdx.x;
    if (b >= B) return;
    const int f = threadIdx.x;                 // 0..255 = feature channel

    if (f < S_NB) sh_idx[f] = neighs[(long)b * S_NB + f];
    __syncthreads();

    const int node = nodes[b];
    const int L    = lens[b];                  // >= 1 by construction

    const float selfv = feats[(long)node * FEAT + f];

    float sum = 0.0f;
    for (int s = 0; s < L; ++s) {
        sum += feats[(long)sh_idx[s] * FEAT + f];
    }

    float* xr = X + (long)b * TWOF;
    xr[f]        = selfv;
    xr[FEAT + f] = sum / (float)L;
}

// ---------------------------------------------------------------------------
// Kernel 2: Y = relu(X @ W^T), X:[B,512], W:[256,512] (both K-contiguous).
// v_wmma_f32_16x16x4_f32; wave tile 16x64, block tile 128x64, K staged in LDS.
// ---------------------------------------------------------------------------
__global__ __launch_bounds__(256)
void sage_gemm_relu(const float* __restrict__ X,
                    const float* __restrict__ W,
                    float*       __restrict__ Y,
                    int B)
{
    __shared__ float Ws[NT * KTP];             // 64 x 66 floats = 16.9 KB

    const int tid     = threadIdx.x;
    const int lane    = tid & 31;
    const int wave    = tid >> 5;              // 0..7
    const int l15     = lane & 15;
    const int halfSel = lane >> 4;             // 0 -> K{0,1}, 1 -> K{2,3}

    const int mWave = blockIdx.x * MT + wave * 16;
    const int nBase = blockIdx.y * NT;

    // A-fragment source row (clamped for the ragged last M tile)
    int rowA = mWave + l15;
    if (rowA > B - 1) rowA = B - 1;
    const float* aRow = X + (long)rowA * TWOF + 2 * halfSel;

    // W staging: thread t -> LDS row n = t>>2, 16 cols at (t&3)*16, as float2s
    const int sn = tid >> 2;                   // 0..63
    const int sc = (tid & 3) * 16;
    const float* wRow  = W + (long)(nBase + sn) * TWOF + sc;
    float*       wsRow = &Ws[sn * KTP + sc];

    v8f acc0 = {}, acc1 = {}, acc2 = {}, acc3 = {};

    for (int k0 = 0; k0 < TWOF; k0 += KT) {
        // ---- stage W[nBase .. nBase+63][k0 .. k0+63] into LDS ----
        #pragma unroll
        for (int j = 0; j < 8; ++j) {
            *(v2f*)(wsRow + j * 2) = *(const v2f*)(wRow + k0 + j * 2);
        }
        __syncthreads();

        // ---- 16 K-steps of 4, 4 WMMAs each ----
        #pragma unroll
        for (int kk = 0; kk < KT; kk += 4) {
            const v2f a = *(const v2f*)(aRow + k0 + kk);
            const float* bp = &Ws[l15 * KTP + kk + 2 * halfSel];
            const v2f b0 = *(const v2f*)(bp);
            const v2f b1 = *(const v2f*)(bp + 16 * KTP);
            const v2f b2 = *(const v2f*)(bp + 32 * KTP);
            const v2f b3 = *(const v2f*)(bp + 48 * KTP);
            acc0 = __builtin_amdgcn_wmma_f32_16x16x4_f32(false, a, false, b0,
                                                         (short)0, acc0, false, false);
            acc1 = __builtin_amdgcn_wmma_f32_16x16x4_f32(false, a, false, b1,
                                                         (short)0, acc1, false, false);
            acc2 = __builtin_amdgcn_wmma_f32_16x16x4_f32(false, a, false, b2,
                                                         (short)0, acc2, false, false);
            acc3 = __builtin_amdgcn_wmma_f32_16x16x4_f32(false, a, false, b3,
                                                         (short)0, acc3, false, false);
        }
        __syncthreads();
    }

    // ---- fused ReLU + store. C/D layout: VGPR r -> M=r (lanes 0-15), M=8+r
    // (lanes 16-31); N = lane&15 within each 16-wide sub-tile. ----
    const int mOff = halfSel * 8;
    const int colB = nBase + l15;
    #pragma unroll
    for (int r = 0; r < 8; ++r) {
        const int row = mWave + mOff + r;
        if (row < B) {
            float* yr = Y + (long)row * OUTF + colB;
            yr[0]  = fmaxf(acc0[r], 0.0f);
            yr[16] = fmaxf(acc1[r], 0.0f);
            yr[32] = fmaxf(acc2[r], 0.0f);
            yr[48] = fmaxf(acc3[r], 0.0f);
        }
    }
}

// ---------------------------------------------------------------------------
extern "C" void kernel_launch(void* const* d_in, const int* in_sizes, int n_in,
                              void* d_out, int out_size, void* d_ws, size_t ws_size,
                              hipStream_t stream)
{
    const float* feats  = (const float*)d_in[0];
    const int*   nodes  = (const int*)  d_in[1];
    const int*   neighs = (const int*)  d_in[2];
    const int*   lens   = (const int*)  d_in[3];
    // d_in[4] = w0: dead in the reference (layer-0 output is overwritten)
    const float* w1     = (const float*)d_in[5];

    const int B = in_sizes[1];                 // 20000 target nodes

    float* X = (float*)d_ws;                   // [B, 512] f32 = 40.96 MB scratch

    sage_prep<<<dim3(B), dim3(256), 0, stream>>>(feats, nodes, neighs, lens, X, B);

    dim3 grid((B + MT - 1) / MT, OUTF / NT);   // (157, 4)
    sage_gemm_relu<<<grid, dim3(256), 0, stream>>>(X, w1, (float*)d_out, B);
}